// ExponentialMatchingLoss_82600811036702
// MI455X (gfx1250) — compile-verified
//
#include <hip/hip_runtime.h>

// ---------------------------------------------------------------------------
// ExponentialMatchingLoss for MI455X (gfx1250, wave32)
//   1) ascending radix sort (4x8-bit LSD, stable via wave32 multisplit)
//   2) two-level PAVA isotonic regression (parallel chunk pooling + serial merge)
//   3) pinball loss reduction
// CDNA5 paths: V_WMMA_F32_16X16X4_F32 for the radix prefix sums (exact for
// integer counts < 2^24), TENSOR_LOAD_TO_LDS (TDM) for scan-tile staging,
// s_wait_tensorcnt, wave32 ballot multisplit, global_prefetch.
// ---------------------------------------------------------------------------

typedef __attribute__((ext_vector_type(2))) float v2f;
typedef __attribute__((ext_vector_type(8))) float v8f;
typedef unsigned int u32x4 __attribute__((ext_vector_type(4)));
typedef int i32x4 __attribute__((ext_vector_type(4)));
typedef int i32x8 __attribute__((ext_vector_type(8)));

#define RADIX_BLOCKS    256
#define WAVES_PER_BLOCK 8
#define NWAVECHUNK      (RADIX_BLOCKS * WAVES_PER_BLOCK)   // 2048 wave-chunks
#define HIST_BINS       256
#define SCAN_M          (HIST_BINS * NWAVECHUNK)           // 524288 scan entries
#define SCAN_BLOCKS     (SCAN_M / 256)                     // 2048
#define PAVA_CHUNK      1024

// ----- float <-> radix-sortable key ----------------------------------------
__device__ __forceinline__ unsigned f2key(float f) {
    unsigned u = __float_as_uint(f);
    return (u & 0x80000000u) ? ~u : (u | 0x80000000u);
}
__device__ __forceinline__ float key2f(unsigned k) {
    unsigned u = (k & 0x80000000u) ? (k & 0x7FFFFFFFu) : ~k;
    return __uint_as_float(u);
}

// ---------------------------------------------------------------------------
// Tensor Data Mover: DMA a contiguous run of `elems` 4-byte words from global
// memory into LDS at byte offset `ldsOff` (1-row 2D descriptor, ISA ch.8).
// D# group0: {count=1 | lds_addr | global_addr[56:0] | type=2}
// D# group1: {data_size=4B, tensor_dim0=tile_dim0=elems, tensor_dim1=1,
//             tensor_dim0_stride=elems}; groups 2/3 zero (<=2D tensor).
// ---------------------------------------------------------------------------
__device__ __forceinline__ void tdm_load_words_to_lds(unsigned ldsOff,
                                                      const void* gsrc,
                                                      unsigned elems) {
    const unsigned long long ga = (unsigned long long)(size_t)gsrc;
    u32x4 g0;
    g0[0] = 1u;                                            // count=1 (user D#)
    g0[1] = ldsOff;                                        // lds_addr (bytes)
    g0[2] = (unsigned)(ga & 0xFFFFFFFFu);                  // global_addr[31:0]
    g0[3] = (unsigned)((ga >> 32) & 0x01FFFFFFu)           // global_addr[56:32]
          | (2u << 30);                                    // type = 2 ("image")
    i32x8 g1;
    g1[0] = (int)(2u << 16);                               // data_size = 4 bytes
    g1[1] = (int)(elems << 16);                            // tensor_dim0[15:0]
    g1[2] = (int)(1u << 16);                               // tensor_dim1 = 1
    g1[3] = (int)(elems << 16);                            // tile_dim0 = elems
    g1[4] = 0;                                             // tile_dim1/2 unused
    g1[5] = (int)elems;                                    // tensor_dim0_stride
    g1[6] = 0;
    g1[7] = 0;
    i32x4 gz4 = {0, 0, 0, 0};
#if defined(__clang_major__) && (__clang_major__ >= 23)
    i32x8 gz8 = {0, 0, 0, 0, 0, 0, 0, 0};
    __builtin_amdgcn_tensor_load_to_lds(g0, g1, gz4, gz4, gz8, 0);
#else
    __builtin_amdgcn_tensor_load_to_lds(g0, g1, gz4, gz4, 0);
#endif
}

// ---------------------------------------------------------------------------
// WMMA-based inclusive prefix sum of 256 f32 values held in LDS (one wave).
// Element e = c*16 + r maps to matrix X[r][c] (columns are contiguous 16-runs).
//   S1 = L @ X          (within-column inclusive scans; L lower-tri ones)
//   P  = colsum @ U     (carry of preceding column totals; U strict upper ones)
//   incl = S1 + P
// Each 16x16x16 product = 4 chained V_WMMA_F32_16X16X4_F32.
// Lane l owns elements e(l,i) = (l&15)*16 + ((l>=16)?8:0) + i, i=0..7.
// ---------------------------------------------------------------------------
__device__ __forceinline__ v8f wmma_scan256_incl(const float* lds, float* col16) {
    const int lane  = threadIdx.x & 31;
    const int nm    = lane & 15;                 // A-row m == B/C-col n
    const int hiOff = (lane & 16) ? 2 : 0;       // K placement for hi half-wave

    v8f s1 = {};
#pragma unroll
    for (int kb = 0; kb < 4; ++kb) {
        const int k0 = kb * 4 + hiOff, k1 = k0 + 1;
        v2f a, b;
        a.x = (k0 <= nm) ? 1.0f : 0.0f;          // L[m][k]
        a.y = (k1 <= nm) ? 1.0f : 0.0f;
        b.x = lds[nm * 16 + k0];                 // X[k][n]
        b.y = lds[nm * 16 + k1];
        s1 = __builtin_amdgcn_wmma_f32_16x16x4_f32(
                 false, a, false, b, (short)0, s1, false, false);
    }

    // Column totals = row 15 of S1 (VGPR 7 of hi lanes) -> LDS broadcast.
    if (lane >= 16) col16[nm] = s1[7];
    __builtin_amdgcn_wave_barrier();

    v8f p = {};
#pragma unroll
    for (int kb = 0; kb < 4; ++kb) {
        const int k0 = kb * 4 + hiOff, k1 = k0 + 1;
        v2f a, b;
        a.x = col16[k0];                         // T[m][k] = colsum[k]
        a.y = col16[k1];
        b.x = (k0 < nm) ? 1.0f : 0.0f;           // U[k][n] strict upper
        b.y = (k1 < nm) ? 1.0f : 0.0f;
        p = __builtin_amdgcn_wmma_f32_16x16x4_f32(
                false, a, false, b, (short)0, p, false, false);
    }
    return s1 + p;
}

// ----- kernel: float -> key ------------------------------------------------
__global__ void k_transform(const float* __restrict__ x, unsigned* __restrict__ keys, int n) {
    for (int i = blockIdx.x * blockDim.x + threadIdx.x; i < n; i += gridDim.x * blockDim.x)
        keys[i] = f2key(x[i]);
}

// ----- kernel: per-wave-chunk 256-bin histogram (bin-major global layout) ---
__global__ void k_hist(const unsigned* __restrict__ keys, unsigned* __restrict__ gHist,
                       int shift, int n, int chunk) {
    __shared__ unsigned sh[WAVES_PER_BLOCK][HIST_BINS];
    const int wave = threadIdx.x >> 5, lane = threadIdx.x & 31;
    for (int i = threadIdx.x; i < WAVES_PER_BLOCK * HIST_BINS; i += blockDim.x)
        ((unsigned*)sh)[i] = 0u;
    __syncthreads();

    const int W    = blockIdx.x * WAVES_PER_BLOCK + wave;
    const int base = W * chunk;
    const int end  = min(n, base + chunk);
    for (int i = base + lane; i < end; i += 32) {
        unsigned bin = (keys[i] >> shift) & 255u;
        atomicAdd(&sh[wave][bin], 1u);
    }
    __syncthreads();
    for (int i = threadIdx.x; i < WAVES_PER_BLOCK * HIST_BINS; i += blockDim.x) {
        int w = i >> 8, b = i & 255;
        gHist[(size_t)b * NWAVECHUNK + (size_t)(blockIdx.x * WAVES_PER_BLOCK + w)] = sh[w][b];
    }
}

// ----- kernel: per-256-tile totals ------------------------------------------
__global__ void k_scan_part(const unsigned* __restrict__ gHist, float* __restrict__ partials) {
    __shared__ float s[256];
    const int t = threadIdx.x;
    s[t] = (float)gHist[(size_t)blockIdx.x * 256 + t];
    __syncthreads();
    for (int o = 128; o > 0; o >>= 1) {
        if (t < o) s[t] += s[t + o];
        __syncthreads();
    }
    if (t == 0) partials[blockIdx.x] = s[0];
}

// ----- kernel: spine — exclusive scan of 2048 tile totals ------------------
// TDM stages all 8KB of spine input into LDS; 8 waves each run a WMMA scan.
__global__ void k_spine(const float* __restrict__ partials, float* __restrict__ spineEx) {
    __shared__ float sd[2048];                   // largest LDS object -> offset 0
    __shared__ float sc[WAVES_PER_BLOCK][16];
    __shared__ float stot[WAVES_PER_BLOCK];
    __shared__ float stotEx[WAVES_PER_BLOCK];

    const int t = threadIdx.x;
    const int wave = t >> 5, lane = t & 31;
    const int sliceBase = wave * 256;
    const int e0 = (lane & 15) * 16 + ((lane >> 4) & 1) * 8;

    if (t == 0) {
        tdm_load_words_to_lds(/*ldsOff(sd)=*/0u, partials, SCAN_BLOCKS);
        __builtin_amdgcn_s_wait_tensorcnt(0);
    }
    __syncthreads();

    float orig[8];
#pragma unroll
    for (int i = 0; i < 8; ++i) orig[i] = sd[sliceBase + e0 + i];

    v8f incl = wmma_scan256_incl(sd + sliceBase, sc[wave]);

    if (lane == 31) stot[wave] = incl[7];   // element 255 of slice = slice total
    __syncthreads();
    if (t == 0) {
        float r = 0.0f;
        for (int w = 0; w < WAVES_PER_BLOCK; ++w) { stotEx[w] = r; r += stot[w]; }
    }
    __syncthreads();
#pragma unroll
    for (int i = 0; i < 8; ++i)
        spineEx[sliceBase + e0 + i] = incl[i] - orig[i] + stotEx[wave];
}

// ----- kernel: per-tile exclusive scan + spine offset (TDM + WMMA) ----------
__global__ void k_scan_apply(unsigned* __restrict__ gHist, const float* __restrict__ spineEx) {
    __shared__ float sd[256];                    // largest LDS object -> offset 0
    __shared__ float sc[16];
    const int lane = threadIdx.x;                // blockDim == 32
    const size_t base = (size_t)blockIdx.x * 256;
    const int e0 = (lane & 15) * 16 + ((lane >> 4) & 1) * 8;

    // DMA the 1KB tile (u32 counts reinterpreted bitwise; consumed as floats
    // after integer->float conversion below).
    tdm_load_words_to_lds(/*ldsOff(sd)=*/0u, gHist + base, 256u);
    __builtin_amdgcn_s_wait_tensorcnt(0);
    __builtin_amdgcn_wave_barrier();

    // Convert the raw u32 counts (landed in LDS) to f32 in place.
    float orig[8];
#pragma unroll
    for (int i = 0; i < 8; ++i) orig[i] = (float)((unsigned*)sd)[e0 + i];
    __builtin_amdgcn_wave_barrier();
#pragma unroll
    for (int i = 0; i < 8; ++i) sd[e0 + i] = orig[i];
    __builtin_amdgcn_wave_barrier();

    v8f incl = wmma_scan256_incl(sd, sc);
    const float off = spineEx[blockIdx.x];
#pragma unroll
    for (int i = 0; i < 8; ++i)
        gHist[base + e0 + i] = (unsigned)(incl[i] - orig[i] + off);
}

// ----- kernel: stable scatter (wave32 multisplit) ---------------------------
__global__ void k_scatter(const unsigned* __restrict__ src, unsigned* __restrict__ dst,
                          const unsigned* __restrict__ gOff, int shift, int n, int chunk) {
    __shared__ unsigned soff[WAVES_PER_BLOCK][HIST_BINS];
    const int wave = threadIdx.x >> 5, lane = threadIdx.x & 31;
    const int W = blockIdx.x * WAVES_PER_BLOCK + wave;

    for (int b = lane; b < HIST_BINS; b += 32)
        soff[wave][b] = gOff[(size_t)b * NWAVECHUNK + W];

    const int base  = W * chunk;
    const int end   = min(n, base + chunk);
    const int iters = (chunk + 31) >> 5;
    for (int it = 0; it < iters; ++it) {
        const int i = base + it * 32 + lane;
        const bool ok = (i < end);
        const unsigned key = ok ? src[i] : 0u;
        const unsigned bin = (key >> shift) & 255u;

        unsigned active = (unsigned)__ballot(ok);
        unsigned mask = active;
#pragma unroll
        for (int b = 0; b < 8; ++b) {
            unsigned bb = (unsigned)__ballot((bin >> b) & 1u);
            mask &= ((bin >> b) & 1u) ? bb : ~bb;
        }
        const int rank   = __popc(mask & ((1u << lane) - 1u));
        const int leader = mask ? (__ffs(mask) - 1) : 0;
        unsigned old = 0u;
        if (ok && lane == leader) old = atomicAdd(&soff[wave][bin], (unsigned)__popc(mask));
        old = __shfl(old, leader);
        if (ok) dst[old + rank] = key;
    }
}

// ----- kernel: per-chunk PAVA (one thread per contiguous chunk) -------------
__global__ void k_pava_chunk(const unsigned* __restrict__ skeys,
                             float* __restrict__ bvals, int* __restrict__ bcnts,
                             int* __restrict__ used, int n, int nChunks) {
    const int c = blockIdx.x * blockDim.x + threadIdx.x;
    if (c >= nChunks) return;
    const int beg = c * PAVA_CHUNK;
    const int end = min(n, beg + PAVA_CHUNK);
    float* vals = bvals + (size_t)c * PAVA_CHUNK;
    int*   cnts = bcnts + (size_t)c * PAVA_CHUNK;
    int top = 0;
    for (int i = beg; i < end; ++i) {
        __builtin_prefetch(skeys + i + 64, 0, 1);       // global_prefetch_b8
        const float xv = key2f(skeys[i]);
        float v = (float)(n - i) * 10.0f + xv;          // y = w + x_asc, w=(n-i)/0.1
        int cc = 1;
        while (top >= 1 && v > vals[top - 1]) {         // strict violator -> pool
            const float c1 = (float)cnts[top - 1];
            v = (vals[top - 1] * c1 + v * (float)cc) / (c1 + (float)cc);
            cc += cnts[top - 1];
            --top;
        }
        vals[top] = v; cnts[top] = cc; ++top;
    }
    used[c] = top;
}

// ----- kernel: serial merge of chunk pools (inherently sequential) ----------
__global__ void k_pava_merge(const float* __restrict__ bvals, const int* bcntsIn,
                             const int* __restrict__ used, int nChunks,
                             float* __restrict__ mvals, int* __restrict__ mstarts,
                             int* __restrict__ nmOut) {
    if (threadIdx.x != 0 || blockIdx.x != 0) return;
    int nm = 0;
    for (int c = 0; c < nChunks; ++c) {
        const int u = used[c];
        const float* vs = bvals   + (size_t)c * PAVA_CHUNK;
        const int*   cs = bcntsIn + (size_t)c * PAVA_CHUNK;
        for (int j = 0; j < u; ++j) {
            float v = vs[j];
            int cc = cs[j];
            while (nm >= 1 && v > mvals[nm - 1]) {
                const float c1 = (float)mstarts[nm - 1];   // counts parked here
                v = (mvals[nm - 1] * c1 + v * (float)cc) / (c1 + (float)cc);
                cc += mstarts[nm - 1];
                --nm;
            }
            mvals[nm] = v; mstarts[nm] = cc; ++nm;         // write idx <= read idx
        }
    }
    int run = 0;                                           // counts -> starts
    for (int b = 0; b < nm; ++b) { const int t = mstarts[b]; mstarts[b] = run; run += t; }
    mstarts[nm] = run;
    *nmOut = nm;
}

// ----- kernel: expand block means to per-element fitted values --------------
__global__ void k_expand(const float* __restrict__ mvals, const int* __restrict__ mstarts,
                         const int* __restrict__ nmPtr, float* __restrict__ fit) {
    const int nm = *nmPtr;
    for (int b = blockIdx.x * blockDim.x + threadIdx.x; b < nm; b += gridDim.x * blockDim.x) {
        const float v = mvals[b];
        const int s = mstarts[b], e = mstarts[b + 1];
        for (int i = s; i < e; ++i) fit[i] = v;
    }
}

// ----- kernel: pinball-loss mean (w terms cancel in the difference) ---------
__global__ void k_loss(const float* __restrict__ fx, const float* __restrict__ fe,
                       float* __restrict__ out, int n) {
    __shared__ float red[256];
    const int tid = threadIdx.x;
    float acc = 0.0f;
    for (int i = blockIdx.x * blockDim.x + tid; i < n; i += gridDim.x * blockDim.x) {
        const float d = fx[i] - fe[i];
        acc += fmaxf(-0.1f * d, 0.9f * d);
    }
    red[tid] = acc;
    __syncthreads();
    for (int o = 128; o > 0; o >>= 1) {
        if (tid < o) red[tid] += red[tid + o];
        __syncthreads();
    }
    if (tid == 0) atomicAdd(out, red[0] / (float)n);
}

// ---------------------------------------------------------------------------
extern "C" void kernel_launch(void* const* d_in, const int* in_sizes, int n_in,
                              void* d_out, int out_size, void* d_ws, size_t ws_size,
                              hipStream_t stream) {
    const float* x  = (const float*)d_in[0];
    const float* xe = (const float*)d_in[1];
    const int n = in_sizes[0];

    char* p = (char*)d_ws;
    auto take = [&](size_t bytes) -> char* {
        char* r = p;
        p += (bytes + 255) & ~(size_t)255;
        return r;
    };
    const size_t nb = (size_t)n * 4;
    unsigned* keyA     = (unsigned*)take(nb);
    unsigned* keyB     = (unsigned*)take(nb);
    float*    bvals    = (float*)take(nb);
    int*      bcnts    = (int*)take(nb);
    float*    fitX     = (float*)take(nb);
    float*    fitE     = (float*)take(nb);
    unsigned* gHist    = (unsigned*)take((size_t)SCAN_M * 4);
    float*    partials = (float*)take((size_t)SCAN_BLOCKS * 4);
    float*    spineEx  = (float*)take((size_t)SCAN_BLOCKS * 4);
    int*      used     = (int*)take(8192 * 4);
    int*      nmPtr    = (int*)take(256);

    const int chunk = (n + NWAVECHUNK - 1) / NWAVECHUNK;
    const int CH    = (n + PAVA_CHUNK - 1) / PAVA_CHUNK;

    for (int arr = 0; arr < 2; ++arr) {
        const float* in = arr ? xe : x;
        float* fit = arr ? fitE : fitX;

        k_transform<<<2048, 256, 0, stream>>>(in, keyA, n);
        for (int pass = 0; pass < 4; ++pass) {
            unsigned* s = (pass & 1) ? keyB : keyA;
            unsigned* d = (pass & 1) ? keyA : keyB;
            k_hist<<<RADIX_BLOCKS, 256, 0, stream>>>(s, gHist, pass * 8, n, chunk);
            k_scan_part<<<SCAN_BLOCKS, 256, 0, stream>>>(gHist, partials);
            k_spine<<<1, 256, 0, stream>>>(partials, spineEx);
            k_scan_apply<<<SCAN_BLOCKS, 32, 0, stream>>>(gHist, spineEx);
            k_scatter<<<RADIX_BLOCKS, 256, 0, stream>>>(s, d, gHist, pass * 8, n, chunk);
        }
        // sorted keys in keyA; keyB is dead -> reuse as merged block values
        k_pava_chunk<<<(CH + 255) / 256, 256, 0, stream>>>(keyA, bvals, bcnts, used, n, CH);
        k_pava_merge<<<1, 32, 0, stream>>>(bvals, bcnts, used, CH,
                                           (float*)keyB, bcnts, nmPtr);
        k_expand<<<1024, 256, 0, stream>>>((const float*)keyB, bcnts, nmPtr, fit);
    }

    hipMemsetAsync(d_out, 0, sizeof(float), stream);
    k_loss<<<1024, 256, 0, stream>>>(fitX, fitE, (float*)d_out, n);
}